// VectorQuantizer_78993038508132
// MI455X (gfx1250) — compile-verified
//
#include <hip/hip_runtime.h>

typedef __attribute__((ext_vector_type(2))) float v2f;
typedef __attribute__((ext_vector_type(4))) float v4f;
typedef __attribute__((ext_vector_type(8))) float v8f;

#define N_CODES   512
#define CODE_DIM  64
#define HW        4096      // h*w
#define B_STRIDE  262144    // c*h*w
#define ROWS_PER_WG 128
#define THREADS   256

extern "C" __global__ __launch_bounds__(THREADS)
void vq_argmin_wmma(const float* __restrict__ x,   // (16,64,64,64) in (b,c,h,w)
                    const float* __restrict__ cb,  // (512,64)
                    float* __restrict__ out)       // flat (N,64) == raw reshape target
{
    extern __shared__ float smem[];
    float* lds_cb = smem;                       // 512*64 floats = 128 KB
    float* lds_cn = smem + N_CODES * CODE_DIM;  // 512 floats

    const int tid = threadIdx.x;

    // ---- stage codebook into LDS (float4, fully coalesced) ----
    {
        const v4f* src = (const v4f*)cb;
        v4f*       dst = (v4f*)lds_cb;
        #pragma unroll
        for (int i = 0; i < (N_CODES * CODE_DIM / 4) / THREADS; ++i)
            dst[tid + i * THREADS] = src[tid + i * THREADS];
    }
    __syncthreads();

    // ---- per-code squared norms ----
    for (int c = tid; c < N_CODES; c += THREADS) {
        const float* row = lds_cb + c * CODE_DIM;
        float s = 0.f;
        #pragma unroll
        for (int d = 0; d < CODE_DIM; ++d) s += row[d] * row[d];
        lds_cn[c] = s;
    }
    __syncthreads();

    const int lane  = tid & 31;
    const int wave  = tid >> 5;
    const int mLane = lane & 15;          // row (A) / code (B,C) within tile
    const int kHalf = (lane >> 4) << 1;   // lanes 0-15: K={0,1}; 16-31: K={2,3}

    const int rowBase = blockIdx.x * ROWS_PER_WG + wave * 16;
    const int bImg    = rowBase >> 12;    // / 4096  (16 rows never straddle a batch)
    const int rem     = rowBase & 4095;

    // ---- A fragments: 16 rows x 64 dims, strided gather from (b,c,h,w) ----
    const float* aptr = x + (size_t)bImg * B_STRIDE + rem + mLane;
    v2f aFrag[16];
    #pragma unroll
    for (int kk = 0; kk < 16; ++kk) {
        const int d0 = kk * 4 + kHalf;
        v2f a;
        a[0] = aptr[(size_t)d0 * HW];
        a[1] = aptr[(size_t)(d0 + 1) * HW];
        aFrag[kk] = a;
    }

    // running argmin: C/D layout -> VGPR r holds row M=r (lanes 0-15) / M=r+8 (16-31)
    float bestVal[8];
    int   bestIdx[8];
    #pragma unroll
    for (int r = 0; r < 8; ++r) { bestVal[r] = __builtin_inff(); bestIdx[r] = 0; }

    // two code tiles per iteration -> two independent WMMA accumulator chains
    for (int t = 0; t < N_CODES / 32; ++t) {
        const int codeBase0 = t * 32;
        const int codeBase1 = codeBase0 + 16;
        const float* bptr0 = lds_cb + (codeBase0 + mLane) * CODE_DIM + kHalf;
        const float* bptr1 = bptr0 + 16 * CODE_DIM;
        v8f acc0 = {0.f, 0.f, 0.f, 0.f, 0.f, 0.f, 0.f, 0.f};
        v8f acc1 = {0.f, 0.f, 0.f, 0.f, 0.f, 0.f, 0.f, 0.f};
        #pragma unroll
        for (int kk = 0; kk < 16; ++kk) {
            v2f b0 = *(const v2f*)(bptr0 + kk * 4);   // ds_load_b64
            v2f b1 = *(const v2f*)(bptr1 + kk * 4);
            acc0 = __builtin_amdgcn_wmma_f32_16x16x4_f32(
                false, aFrag[kk], false, b0, (short)0, acc0, false, false);
            acc1 = __builtin_amdgcn_wmma_f32_16x16x4_f32(
                false, aFrag[kk], false, b1, (short)0, acc1, false, false);
        }
        const float cn0 = lds_cn[codeBase0 + mLane];
        const float cn1 = lds_cn[codeBase1 + mLane];
        const int   idx0 = codeBase0 + mLane;
        const int   idx1 = codeBase1 + mLane;
        #pragma unroll
        for (int r = 0; r < 8; ++r) {
            const float s0 = cn0 - 2.f * acc0[r];   // ||c||^2 - 2 x.c  (tile 0 first:
            if (s0 < bestVal[r]) { bestVal[r] = s0; bestIdx[r] = idx0; } // lower idx)
            const float s1 = cn1 - 2.f * acc1[r];
            if (s1 < bestVal[r]) { bestVal[r] = s1; bestIdx[r] = idx1; }
        }
    }

    // ---- butterfly argmin across the 16 lanes of each half (first-index ties) ----
    #pragma unroll
    for (int r = 0; r < 8; ++r) {
        float v = bestVal[r]; int i = bestIdx[r];
        #pragma unroll
        for (int m = 1; m <= 8; m <<= 1) {
            float ov = __shfl_xor(v, m, 32);
            int   oi = __shfl_xor(i, m, 32);
            if (ov < v || (ov == v && oi < i)) { v = ov; i = oi; }
        }
        bestIdx[r] = i;
    }

    // ---- write 16 rows: whole wave streams one code row per iteration ----
    #pragma unroll
    for (int m = 0; m < 16; ++m) {
        const int r       = m & 7;
        const int srcLane = (m >> 3) << 4;              // 0 -> rows 0-7, 16 -> rows 8-15
        const int idx     = __shfl(bestIdx[r], srcLane, 32);
        const float* crow = lds_cb + idx * CODE_DIM + lane * 2;
        float*       orow = out + (size_t)(rowBase + m) * CODE_DIM + lane * 2;
        *(v2f*)orow = *(const v2f*)crow;                // 256B coalesced store
    }
}

extern "C" void kernel_launch(void* const* d_in, const int* in_sizes, int n_in,
                              void* d_out, int out_size, void* d_ws, size_t ws_size,
                              hipStream_t stream) {
    const float* x   = (const float*)d_in[0];
    const float* cb  = (const float*)d_in[1];
    float*       out = (float*)d_out;

    const int rows = in_sizes[0] / CODE_DIM;        // 65536
    const int nwg  = rows / ROWS_PER_WG;            // 512
    const size_t smem = (size_t)(N_CODES * CODE_DIM + N_CODES) * sizeof(float); // 133120 B

    hipLaunchKernelGGL(vq_argmin_wmma, dim3(nwg), dim3(THREADS), smem, stream,
                       x, cb, out);
}